// LsunBlockIdct2dLayer_73959336837579
// MI455X (gfx1250) — compile-verified
//
#include <hip/hip_runtime.h>
#include <hip/hip_bf16.h>

typedef __attribute__((ext_vector_type(2))) float v2f;
typedef __attribute__((ext_vector_type(4))) float v4f;
typedef __attribute__((ext_vector_type(8))) float v8f;

// Orthonormal 4-point DCT-II matrix entries (all compile-time constants):
//   C0[t] = H,  C1[t] = { A, B, -B, -A },  C2[t] = { H, -H, -H, H },
//   C3[t] = { B, -A, A, -B };  H = 0.5, A = cos(pi/8)/sqrt2, B = cos(3pi/8)/sqrt2
#define H_ 0.5f
#define A_ 0.65328148f
#define B_ 0.27059805f
#define C1_(t) ((t) == 0 ? A_ : (t) == 1 ? B_ : (t) == 2 ? -B_ : -A_)
#define C2_(t) (((t) == 1 || (t) == 2) ? -H_ : H_)
#define C3_(t) ((t) == 0 ? B_ : (t) == 1 ? -A_ : (t) == 2 ? A_ : -B_)

// Per-lane A-operand row of the fused (permute + 2D-IDCT) operator M',
// laid out exactly as WMMA_16x16x4 A-fragments for the 4 chained WMMAs:
//   { a0.x, a0.y, a1.x, a1.y, a2.x, a2.y, a3.x, a3.y }
// lane = h*16 + m, m = 4p+q;  cie = C[2h][p], cio = C[2h+1][p].
#define ROW(cie, cio, q)                                      \
  (cie) * H_, (cie) * C2_(q), (cio) * C1_(q), (cio) * C3_(q), \
  (cio) * H_, (cio) * C2_(q), (cie) * C1_(q), (cie) * C3_(q)
#define P4(cie, cio) \
  ROW(cie, cio, 0), ROW(cie, cio, 1), ROW(cie, cio, 2), ROW(cie, cio, 3)

// 32 lanes x 8 floats = 1KB, constant-folded at compile time; permanently
// cache-resident (every wave reads the same 1KB, never evicted by NT streams).
__device__ const float AMAT[256] = {
    // h = 0 (lanes 0-15): cie = C0[p] = H, cio = C1[p]
    P4(H_, C1_(0)), P4(H_, C1_(1)), P4(H_, C1_(2)), P4(H_, C1_(3)),
    // h = 1 (lanes 16-31): cie = C2[p], cio = C3[p]
    P4(C2_(0), C3_(0)), P4(C2_(1), C3_(1)), P4(C2_(2), C3_(2)), P4(C2_(3), C3_(3)),
};

// One wave handles 4 groups of 16 blocks (8KB streamed). Per group:
//   D(16x16) = M'(16x16) @ X(16 coefs x 16 blocks) via 4x WMMA_F32_16X16X4_F32.
// Output flat layout == input flat layout (the reference's reshape is flat),
// so both sides are contiguous 64B/block; every wave touches contiguous 4KB
// in and 4KB out. Memory-bound: ~268MB total -> ~11.5us at 23.3 TB/s.
__global__ __launch_bounds__(256) void lsun_idct4x4_wmma(
    const float* __restrict__ x, float* __restrict__ y, unsigned nblk) {
  const unsigned lane = threadIdx.x & 31u;
  const unsigned wid = (blockIdx.x * blockDim.x + threadIdx.x) >> 5;
  const unsigned g0 = wid * 64u;  // first of this wave's 64 blocks
  if (g0 >= nblk) return;         // wave-uniform exit: EXEC all-ones for WMMA

  const unsigned n = lane & 15u;  // block within group / D column
  const unsigned h = lane >> 4;   // half-wave selector

  // ---- A operands: 2x global_load_b128 from the hot 1KB constant table ----
  const v4f* ap = (const v4f*)(AMAT + lane * 8u);
  const v4f alo = ap[0];
  const v4f ahi = ap[1];
  const v2f a0 = alo.lo, a1 = alo.hi, a2 = ahi.lo, a3 = ahi.hi;

  // Lane's byte offsets; all loop iterations fold into immediate offsets.
  const char* ib = (const char*)x + (g0 + n) * 64u + h * 8u;
  char* ob = (char*)y + (g0 + n) * 64u + h * 32u;

#pragma unroll
  for (int it = 0; it < 4; ++it) {
    // B operands: coef pairs (4j+2h, 4j+1+2h) of block n of group it
    const char* ip = ib + it * 1024;
    const v2f b0 = __builtin_nontemporal_load((const v2f*)(ip + 0));
    const v2f b1 = __builtin_nontemporal_load((const v2f*)(ip + 16));
    const v2f b2 = __builtin_nontemporal_load((const v2f*)(ip + 32));
    const v2f b3 = __builtin_nontemporal_load((const v2f*)(ip + 48));

    // 4 chained WMMAs accumulate D = M' @ X over K = 16
    v8f acc = {0.f, 0.f, 0.f, 0.f, 0.f, 0.f, 0.f, 0.f};
    acc = __builtin_amdgcn_wmma_f32_16x16x4_f32(false, a0, false, b0, (short)0,
                                                acc, false, false);
    acc = __builtin_amdgcn_wmma_f32_16x16x4_f32(false, a1, false, b1, (short)0,
                                                acc, false, false);
    acc = __builtin_amdgcn_wmma_f32_16x16x4_f32(false, a2, false, b2, (short)0,
                                                acc, false, false);
    acc = __builtin_amdgcn_wmma_f32_16x16x4_f32(false, a3, false, b3, (short)0,
                                                acc, false, false);

    // D layout: lane holds outputs m = 8h..8h+7 of block n -> contiguous 32B
    char* op = ob + it * 1024;
    __builtin_nontemporal_store(acc.lo, (v4f*)(op + 0));
    __builtin_nontemporal_store(acc.hi, (v4f*)(op + 16));
  }
}

extern "C" void kernel_launch(void* const* d_in, const int* in_sizes, int n_in,
                              void* d_out, int out_size, void* d_ws,
                              size_t ws_size, hipStream_t stream) {
  (void)n_in; (void)out_size; (void)d_ws; (void)ws_size;
  const float* x = (const float*)d_in[0];
  float* y = (float*)d_out;

  const unsigned nblk = (unsigned)(in_sizes[0] / 16);   // 2,097,152 blocks
  const unsigned nwaves = (nblk + 63u) / 64u;           // 64 blocks per wave
  const int threads = 256;                              // 8 waves per WG
  const long long total_threads = (long long)nwaves * 32;
  const int grid = (int)((total_threads + threads - 1) / threads);

  lsun_idct4x4_wmma<<<grid, threads, 0, stream>>>(x, y, nblk);
}